// NNstack_60816736911761
// MI455X (gfx1250) — compile-verified
//
#include <hip/hip_runtime.h>

// Problem shape (fixed by the reference's setup_inputs)
#define T_PREV 1023
#define T_TOT  1024
#define BATCH  512
#define MDIM   128

#define NCHUNK 8          // t-chunks per batch (partials)
#define WAVES  8          // waves per block (256 threads, wave32)
#define ROWS_PER_WAVE 16  // t-rows per wave => 8*16 = 128 rows per block

// gfx12-family cache policy immediates: TH in bits [2:0] (0=RT, 1=NT)
#define CPOL_RT 0
#define CPOL_NT 1

// ---- CDNA5 feature probes (device pass only) -------------------------------
#if defined(__gfx1250__) && __has_builtin(__builtin_amdgcn_wmma_f32_16x16x4_f32)
#define HAVE_WMMA 1
#else
#define HAVE_WMMA 0
#endif

#if defined(__gfx1250__) && \
    __has_builtin(__builtin_amdgcn_global_load_async_to_lds_b128) && \
    __has_builtin(__builtin_amdgcn_global_store_async_from_lds_b128) && \
    __has_builtin(__builtin_amdgcn_s_wait_asynccnt)
#define HAVE_ASYNC 1
#else
#define HAVE_ASYNC 0
#endif

typedef float v2f __attribute__((ext_vector_type(2)));
typedef float v8f __attribute__((ext_vector_type(8)));
// The async builtins take GCC-style int4 pointers in AS(1)/AS(3).
typedef int v4i_g __attribute__((vector_size(4 * sizeof(int))));

// Generic -> AS(1)/AS(3) pointers via integer casts.
// AMDGPU generic pointers to LDS carry the LDS byte offset in the low 32 bits,
// so truncation yields a valid AS(3) pointer; global pointers share their
// 64-bit representation with generic.
#define AS1PTR(p) ((__attribute__((address_space(1))) v4i_g*)(unsigned long long)(p))
#define AS3PTR(p) ((__attribute__((address_space(3))) v4i_g*)(unsigned int)(unsigned long long)(p))

// ---------------------------------------------------------------------------
// K1: sequential top-down scan per batch. Produces stg (output) and coef (ws).
//   loop1: s = min(prev_stg - min(u,0), 0); u -= s
//   loop2 carry fused in same top-down order: coef = min(s, min(read,0)); read -= s
// First step of loop2 uses (s=dt, v=vt) at t = T_PREV.
// ---------------------------------------------------------------------------
__global__ void __launch_bounds__(64)
scan_kernel(const float* __restrict__ prev_stg,
            const float* __restrict__ dt,
            const float* __restrict__ ut,
            float* __restrict__ stg_out,
            float* __restrict__ coef)
{
    const int b = blockIdx.x * blockDim.x + threadIdx.x;  // 0..511
    float u    = ut[b];
    float read = 1.0f;

    // t = T_PREV row: s = dt[b]
    {
        const float s = dt[b];
        stg_out[(size_t)T_PREV * BATCH + b] = s;
        coef[(size_t)T_PREV * BATCH + b]    = fminf(s, fminf(read, 0.0f));
        read -= s;
    }

#pragma unroll 4
    for (int i = T_PREV - 1; i >= 0; --i) {
        const float sp   = prev_stg[(size_t)i * BATCH + b];
        const float tmp  = sp - fminf(u, 0.0f);
        const float s    = fminf(tmp, 0.0f);
        u -= s;
        stg_out[(size_t)i * BATCH + b] = s;
        coef[(size_t)i * BATCH + b]    = fminf(s, fminf(read, 0.0f));
        read -= s;
    }
}

// ---------------------------------------------------------------------------
// K2: bandwidth kernel. Block = (chunk, b). Each wave stages 16 rows of
// Val[t][b][0:128] into LDS (async DMA on CDNA5, non-temporal: touch-once
// 268 MB stream vs 192 MB L2), copies them out to the Val output (LDS->HBM
// async store, also NT), and accumulates coef[t,b]*row into a partial rt
// using v_wmma_f32_16x16x4_f32 (A = coef broadcast over all 16 rows).
// ---------------------------------------------------------------------------
__global__ void __launch_bounds__(256)
stream_kernel(const float* __restrict__ prev_Val,
              const float* __restrict__ vt,
              const float* __restrict__ coef,
              float* __restrict__ Val_out,
              float* __restrict__ partial)
{
    __shared__ float tile[WAVES][ROWS_PER_WAVE][MDIM];  // 64 KB
    __shared__ float red[WAVES][MDIM];                  //  4 KB

    const int tid   = threadIdx.x;
    const int lane  = tid & 31;
    const int w     = tid >> 5;
    const int b     = blockIdx.y;
    const int chunk = blockIdx.x;
    const int tbase = chunk * (WAVES * ROWS_PER_WAVE) + w * ROWS_PER_WAVE;

    // ---- stage 16 rows into LDS (and emit the Val copy) ----
#if HAVE_ASYNC
#pragma unroll
    for (int r = 0; r < ROWS_PER_WAVE; ++r) {
        const int t = tbase + r;
        const float* src = (t < T_PREV)
            ? (prev_Val + ((size_t)t * BATCH + b) * MDIM)
            : (vt + (size_t)b * MDIM);
        __builtin_amdgcn_global_load_async_to_lds_b128(
            AS1PTR(src + lane * 4), AS3PTR(&tile[w][r][lane * 4]), 0, CPOL_NT);
    }
    __builtin_amdgcn_s_wait_asynccnt(0);
#pragma unroll
    for (int r = 0; r < ROWS_PER_WAVE; ++r) {
        const int t = tbase + r;
        __builtin_amdgcn_global_store_async_from_lds_b128(
            AS1PTR(Val_out + ((size_t)t * BATCH + b) * MDIM + lane * 4),
            AS3PTR(&tile[w][r][lane * 4]), 0, CPOL_NT);
    }
#else
#pragma unroll
    for (int r = 0; r < ROWS_PER_WAVE; ++r) {
        const int t = tbase + r;
        const float* src = (t < T_PREV)
            ? (prev_Val + ((size_t)t * BATCH + b) * MDIM)
            : (vt + (size_t)b * MDIM);
        const float4 v = *(const float4*)(src + lane * 4);
        *(float4*)&tile[w][r][lane * 4] = v;
        *(float4*)(Val_out + ((size_t)t * BATCH + b) * MDIM + lane * 4) = v;
    }
#endif
    __syncthreads();   // make LDS tile visible across lanes (wmma B-operand is cross-lane)

    // ---- reduction: rt_partial[m] = sum_r coef[t,b] * tile[r][m] ----
#if HAVE_WMMA
    // A (16x4 f32) layout: lanes 0-15 hold K=0 (v0), K=1 (v1); lanes 16-31 K=2,3.
    // We broadcast coef over all 16 A-rows, so every row of D is the same
    // partial result; B (4x16) takes Val with k striped the same way.
    v8f acc[8] = {};
    const int kg  = (lane >> 4) << 1;  // 0 or 2
    const int col = lane & 15;
#pragma unroll
    for (int q = 0; q < 4; ++q) {
        const int t0 = tbase + q * 4;
        const float c0 = coef[(size_t)(t0 + 0) * BATCH + b];
        const float c1 = coef[(size_t)(t0 + 1) * BATCH + b];
        const float c2 = coef[(size_t)(t0 + 2) * BATCH + b];
        const float c3 = coef[(size_t)(t0 + 3) * BATCH + b];
        v2f A;
        A[0] = (kg == 0) ? c0 : c2;
        A[1] = (kg == 0) ? c1 : c3;
        // shared base so the two k-rows become a ds_load_2addr pair
        const float* bp = &tile[w][q * 4 + kg][col];
#pragma unroll
        for (int mt = 0; mt < 8; ++mt) {
            v2f B;
            B[0] = bp[mt * 16];
            B[1] = bp[mt * 16 + MDIM];
            acc[mt] = __builtin_amdgcn_wmma_f32_16x16x4_f32(
                false, A, false, B, (short)0, acc[mt], false, false);
        }
    }
    // D rows are identical; lane L (0..15), VGPR0 holds (M=0, N=L) -> m = mt*16+L
    if (lane < 16) {
#pragma unroll
        for (int mt = 0; mt < 8; ++mt)
            red[w][mt * 16 + lane] = acc[mt][0];
    }
#else
    float ax = 0.f, ay = 0.f, az = 0.f, aw4 = 0.f;
#pragma unroll
    for (int r = 0; r < ROWS_PER_WAVE; ++r) {
        const int t = tbase + r;
        const float c = coef[(size_t)t * BATCH + b];
        const float4 v = *(const float4*)&tile[w][r][lane * 4];
        ax += c * v.x; ay += c * v.y; az += c * v.z; aw4 += c * v.w;
    }
    red[w][lane * 4 + 0] = ax;
    red[w][lane * 4 + 1] = ay;
    red[w][lane * 4 + 2] = az;
    red[w][lane * 4 + 3] = aw4;
#endif
    __syncthreads();

    // cross-wave reduce -> per-chunk partial
    if (tid < MDIM) {
        float s = 0.f;
#pragma unroll
        for (int ww = 0; ww < WAVES; ++ww) s += red[ww][tid];
        partial[((size_t)chunk * BATCH + b) * MDIM + tid] = s;
    }
}

// ---------------------------------------------------------------------------
// K3: rt[b,m] = sum over chunks of partial[chunk][b][m]
// ---------------------------------------------------------------------------
__global__ void __launch_bounds__(256)
reduce_kernel(const float* __restrict__ partial, float* __restrict__ rt)
{
    const int idx = blockIdx.x * blockDim.x + threadIdx.x;  // 0 .. B*M-1
    float s = 0.f;
#pragma unroll
    for (int c = 0; c < NCHUNK; ++c)
        s += partial[(size_t)c * BATCH * MDIM + idx];
    rt[idx] = s;
}

// ---------------------------------------------------------------------------
extern "C" void kernel_launch(void* const* d_in, const int* in_sizes, int n_in,
                              void* d_out, int out_size, void* d_ws, size_t ws_size,
                              hipStream_t stream)
{
    (void)in_sizes; (void)n_in; (void)out_size; (void)ws_size;

    const float* prev_Val = (const float*)d_in[0];  // (1023, 512, 128)
    const float* prev_stg = (const float*)d_in[1];  // (1023, 512)
    const float* dt       = (const float*)d_in[2];  // (512,)
    const float* ut       = (const float*)d_in[3];  // (512,)
    const float* vt       = (const float*)d_in[4];  // (1, 512, 128)

    float* Val_out = (float*)d_out;                              // (1024,512,128)
    float* stg_out = Val_out + (size_t)T_TOT * BATCH * MDIM;     // (1024,512)
    float* rt_out  = stg_out + (size_t)T_TOT * BATCH;            // (512,128)

    float* coef    = (float*)d_ws;                               // 1024*512 f32 (2 MB)
    float* partial = coef + (size_t)T_TOT * BATCH;               // 8*512*128 f32 (2 MB)

    scan_kernel<<<dim3(BATCH / 64), dim3(64), 0, stream>>>(prev_stg, dt, ut, stg_out, coef);
    stream_kernel<<<dim3(NCHUNK, BATCH), dim3(256), 0, stream>>>(prev_Val, vt, coef, Val_out, partial);
    reduce_kernel<<<dim3((BATCH * MDIM) / 256), dim3(256), 0, stream>>>(partial, rt_out);
}